// MarginLoss_22187801051754
// MI455X (gfx1250) — compile-verified
//
#include <hip/hip_runtime.h>
#include <hip/hip_bf16.h>

// ---------------------------------------------------------------------------
// MarginLoss reduction for MI455X (gfx1250, wave32).
// Memory-bound: 105 MB streamed once -> ~4.5 us floor at 23.3 TB/s.
//
// Algebraic split to keep the hot loop at 2 VALU/element:
//   L = 0.5 * SUM_all (p-0.1)^2                      (no target, no select)
//     + SUM_rows [ relu(0.9-p_t)^2 - 0.5*(p_t-0.1)^2 ]  (one gather per row)
// The row-gather pass runs FIRST so its cache lines sit in the 192 MB L2 and
// are re-hit by the streaming pass -> no net extra HBM traffic.
//
// Wave reduction via V_WMMA_F32_16X16X4_F32 (A = lane partials, B = ones):
// D[m][n] = x_m + x_{m+16}; lane n sums D rows 0..7, lane n+16 rows 8..15,
// so per-half-wave leaders write LDS and thread 0 folds in fixed order
// (fully deterministic, no float atomics, no shuffles).
// ---------------------------------------------------------------------------

typedef float v2f __attribute__((ext_vector_type(2)));
typedef float v4f __attribute__((ext_vector_type(4)));
typedef float v8f __attribute__((ext_vector_type(8)));

#define M_PLUS_F   0.9f
#define M_MINUS_F  0.1f
#define WEIGHT_F   0.5f
#define BATCH_N    262144
#define NB_CLASSES 100
#define NQ         (BATCH_N * NB_CLASSES / 4)   // 6,553,600 float4s
#define BLOCKS1    2048
#define TPB        256
#define NHALF      (TPB / 16)                   // 16 half-wave slots per block

// Returns, in lane n (n<16): sum_{m=0..7} (x_m + x_{m+16});
//          in lane n+16:     sum_{m=8..15}(x_m + x_{m+16}).
// Caller combines the two half-wave values via LDS.
// EXEC must be all ones (call with the full wave live, no early returns).
__device__ __forceinline__ float halfwave_reduce_wmma(float x) {
    v2f a; a.x = x;    a.y = 0.0f;   // A[m][0]=x (lanes<16), A[m][2]=x (lanes>=16)
    v2f b; b.x = 1.0f; b.y = 1.0f;   // B = ones(4x16)
    v8f c = {};
    v8f d = __builtin_amdgcn_wmma_f32_16x16x4_f32(
        /*neg_a=*/false, a, /*neg_b=*/false, b,
        /*c_mod=*/(short)0, c, /*reuse_a=*/false, /*reuse_b=*/false);
    return d[0] + d[1] + d[2] + d[3] + d[4] + d[5] + d[6] + d[7];
}

__global__ __launch_bounds__(TPB)
void margin_loss_partial(const float* __restrict__ in,
                         const int* __restrict__ tgt32,   // low b32 of int64 targets
                         float* __restrict__ partials) {
    const int tid    = blockIdx.x * TPB + threadIdx.x;
    const int stride = BLOCKS1 * TPB;
    const v4f* in4   = (const v4f*)in;

    // ---- Pass A: per-row correction gather (also pre-warms L2 lines) ----
    float corr = 0.0f;
    for (int r = tid; r < BATCH_N; r += stride) {
        const int   t  = tgt32[2 * r];            // int64 LE low word, class < 100
        const float p  = in[r * NB_CLASSES + t];  // regular (RT) load -> stays in L2
        const float dp = fmaxf(M_PLUS_F - p, 0.0f);
        const float dn = p - M_MINUS_F;
        corr += dp * dp - WEIGHT_F * dn * dn;
    }

    // ---- Pass B: bulk stream, 2 VALU per element (sub + fma) ----
    float raw = 0.0f;   // SUM (p - 0.1)^2 ; the *0.5 is applied once below
    for (int q = tid; q < NQ; q += stride) {
        const v4f v = __builtin_nontemporal_load(in4 + q);  // NT: don't thrash L2
        const float d0 = v.x - M_MINUS_F;
        const float d1 = v.y - M_MINUS_F;
        const float d2 = v.z - M_MINUS_F;
        const float d3 = v.w - M_MINUS_F;
        raw = fmaf(d0, d0, raw);
        raw = fmaf(d1, d1, raw);
        raw = fmaf(d2, d2, raw);
        raw = fmaf(d3, d3, raw);
    }

    const float lane_sum = fmaf(WEIGHT_F, raw, corr);

    // Full block still live here -> EXEC all ones for WMMA.
    const float h = halfwave_reduce_wmma(lane_sum);

    __shared__ float lds[NHALF];
    if ((threadIdx.x & 15) == 0) lds[threadIdx.x >> 4] = h;
    __syncthreads();

    if (threadIdx.x == 0) {
        float b = 0.0f;
        #pragma unroll
        for (int i = 0; i < NHALF; ++i) b += lds[i];   // fixed order: deterministic
        partials[blockIdx.x] = b;
    }
}

__global__ __launch_bounds__(TPB)
void margin_loss_final(const float* __restrict__ partials,
                       float* __restrict__ out) {
    float acc = 0.0f;
    for (int i = threadIdx.x; i < BLOCKS1; i += TPB)   // fixed association
        acc += partials[i];

    const float h = halfwave_reduce_wmma(acc);

    __shared__ float lds[NHALF];
    if ((threadIdx.x & 15) == 0) lds[threadIdx.x >> 4] = h;
    __syncthreads();

    if (threadIdx.x == 0) {
        float tot = 0.0f;
        #pragma unroll
        for (int i = 0; i < NHALF; ++i) tot += lds[i];
        out[0] = tot * (1.0f / (float)BATCH_N);
    }
}

extern "C" void kernel_launch(void* const* d_in, const int* in_sizes, int n_in,
                              void* d_out, int out_size, void* d_ws, size_t ws_size,
                              hipStream_t stream) {
    (void)in_sizes; (void)n_in; (void)out_size; (void)ws_size;

    const float* in    = (const float*)d_in[0];   // f32 [262144,100], row-major
    const int*   tgt32 = (const int*)d_in[1];     // int64 targets, read low words
    float* partials    = (float*)d_ws;            // 2048 floats = 8 KB scratch
    float* out         = (float*)d_out;

    margin_loss_partial<<<BLOCKS1, TPB, 0, stream>>>(in, tgt32, partials);
    margin_loss_final<<<1, TPB, 0, stream>>>(partials, out);
}